// DigitCaps_52132313039396
// MI455X (gfx1250) — compile-verified
//
#include <hip/hip_runtime.h>

// DigitCaps fused kernel for gfx1250 (wave32, WMMA f32 16x16x4).
//
// Shapes: x[B=128, J=4608, 8], W[J, 8, 32], dc[10, 8].
// Votes: u[b, j, m*8+d], n = j*4+m, N = 18432.
// Pass 1 (digitcaps_votes): per j, compute u^T = W_j^T (32x8) * x_j^T (8x16)
//   with V_WMMA_F32_16X16X4_F32. Transposed orientation puts the full vote
//   vector (d=0..7) into one lane's 8 accumulator VGPRs. Lane-local argmax
//   vs dc (LDS broadcast), then LDS-atomic accumulate:
//     s[b][8]  = sum of votes per batch   (for final mean-similarity)
//     seg[c][8], cnt[c]                    (segment-sum update)
// Pass 2 (digitcaps_final): dc_new = dc + (seg - cnt*dc)/(B*N);
//   out[b,c] = <s[b], dc_new[c]> / N; append dc_new.
//
// ws layout (floats): s[1024] @0, seg[80] @1024, cnt[10] @1104 (1114 total).

typedef __attribute__((ext_vector_type(2))) float v2f;
typedef __attribute__((ext_vector_type(8))) float v8f;

#define J_TOT   4608
#define JPB     8          // positions per block
#define XSTRIDE (J_TOT * 8)

__global__ __launch_bounds__(256) void digitcaps_votes(
    const float* __restrict__ x,   // [128][4608][8]
    const float* __restrict__ W,   // [4608][8][32]
    const float* __restrict__ dc,  // [10][8]
    float* __restrict__ ws)
{
    __shared__ float dc_lds[80];
    __shared__ float s_lds[1024];   // [b=128][d=8]
    __shared__ float seg_lds[80];   // [c=10][d=8]
    __shared__ float cnt_lds[10];

    const int t = threadIdx.x;
    if (t < 80) { dc_lds[t] = dc[t]; seg_lds[t] = 0.0f; }
    if (t < 10) cnt_lds[t] = 0.0f;
    for (int i = t; i < 1024; i += 256) s_lds[i] = 0.0f;
    __syncthreads();

    const int wave = t >> 5;          // 0..7 -> batch tile
    const int lane = t & 31;
    const int g    = lane >> 4;       // lane group 0/1
    const int l    = lane & 15;
    const int brow = (wave << 4) + l; // global batch index this lane's column

    for (int jj = 0; jj < JPB; ++jj) {
        const int j = blockIdx.x * JPB + jj;
        const float* Wj = W + (size_t)j * 256;                 // [i=8][dm=32]
        const float* xp = x + (size_t)brow * XSTRIDE + j * 8;  // [i=8]

        // B operand = x_j^T (8x16): lane l = column b, VGPR v -> K = k0 + 2g + v
        v2f bf0, bf1;
        {
            const float2 p0 = *(const float2*)(xp + 0 + 2 * g);
            const float2 p1 = *(const float2*)(xp + 4 + 2 * g);
            bf0.x = p0.x; bf0.y = p0.y;
            bf1.x = p1.x; bf1.y = p1.y;
        }
        // A operand = W_j^T (16x8 per row-tile): M = dm = 16*tile + l,
        // VGPR v -> K(i) = k0 + 2g + v ; A[M][K] = Wj[i*32 + dm]
        const int i0 = (0 + 2 * g) * 32;   // k-step 0
        const int i1 = (4 + 2 * g) * 32;   // k-step 1
        v2f a00, a01, a10, a11;
        a00.x = Wj[i0 + l];       a00.y = Wj[i0 + 32 + l];
        a01.x = Wj[i1 + l];       a01.y = Wj[i1 + 32 + l];
        a10.x = Wj[i0 + 16 + l];  a10.y = Wj[i0 + 32 + 16 + l];
        a11.x = Wj[i1 + 16 + l];  a11.y = Wj[i1 + 32 + 16 + l];

        v8f acc0 = {0.f,0.f,0.f,0.f,0.f,0.f,0.f,0.f};  // dm rows 0..15  (m = g)
        v8f acc1 = {0.f,0.f,0.f,0.f,0.f,0.f,0.f,0.f};  // dm rows 16..31 (m = 2+g)
        acc0 = __builtin_amdgcn_wmma_f32_16x16x4_f32(false, a00, false, bf0,
                                                     (short)0, acc0, false, false);
        acc0 = __builtin_amdgcn_wmma_f32_16x16x4_f32(false, a01, false, bf1,
                                                     (short)0, acc0, false, false);
        acc1 = __builtin_amdgcn_wmma_f32_16x16x4_f32(false, a10, false, bf0,
                                                     (short)0, acc1, false, false);
        acc1 = __builtin_amdgcn_wmma_f32_16x16x4_f32(false, a11, false, bf1,
                                                     (short)0, acc1, false, false);

        // Each lane now owns two complete votes (d = register index 0..7).
        #pragma unroll
        for (int tile = 0; tile < 2; ++tile) {
            v8f v = tile ? acc1 : acc0;
            // argmax_c <v, dc[c]>  (first-max wins, matching jnp.argmax)
            float best = -3.402823466e38f;
            int wc = 0;
            #pragma unroll
            for (int c = 0; c < 10; ++c) {
                float sim = 0.0f;
                #pragma unroll
                for (int r = 0; r < 8; ++r) sim += v[r] * dc_lds[c * 8 + r];
                if (sim > best) { best = sim; wc = c; }
            }
            #pragma unroll
            for (int r = 0; r < 8; ++r) {
                unsafeAtomicAdd(&s_lds[brow * 8 + r], v[r]);
                unsafeAtomicAdd(&seg_lds[wc * 8 + r], v[r]);
            }
            unsafeAtomicAdd(&cnt_lds[wc], 1.0f);
        }
    }

    __syncthreads();
    float* g_s   = ws;
    float* g_seg = ws + 1024;
    float* g_cnt = ws + 1104;
    for (int i = t; i < 1024; i += 256) unsafeAtomicAdd(&g_s[i], s_lds[i]);
    if (t < 80) unsafeAtomicAdd(&g_seg[t], seg_lds[t]);
    if (t < 10) unsafeAtomicAdd(&g_cnt[t], cnt_lds[t]);
}

__global__ __launch_bounds__(128) void digitcaps_final(
    const float* __restrict__ dc,
    const float* __restrict__ ws,
    float* __restrict__ out)   // [128*10 output | 80 dc_new]
{
    __shared__ float dcn[80];
    const int t = threadIdx.x;
    const float* g_s   = ws;
    const float* g_seg = ws + 1024;
    const float* g_cnt = ws + 1104;

    if (t < 80) {
        const int c = t >> 3;
        const float d0 = dc[t];
        // LR*THETA = 1; denom = B*N = 128*18432 = 2359296
        const float nd = d0 + (g_seg[t] - g_cnt[c] * d0) * (1.0f / 2359296.0f);
        dcn[t] = nd;
        out[1280 + t] = nd;
    }
    __syncthreads();

    // t = batch index b
    float sb[8];
    #pragma unroll
    for (int r = 0; r < 8; ++r) sb[r] = g_s[t * 8 + r];
    const float invN = 1.0f / 18432.0f;
    #pragma unroll
    for (int c = 0; c < 10; ++c) {
        float acc = 0.0f;
        #pragma unroll
        for (int r = 0; r < 8; ++r) acc += sb[r] * dcn[c * 8 + r];
        out[t * 10 + c] = acc * invN;
    }
}

extern "C" void kernel_launch(void* const* d_in, const int* in_sizes, int n_in,
                              void* d_out, int out_size, void* d_ws, size_t ws_size,
                              hipStream_t stream) {
    const float* x  = (const float*)d_in[0];   // inputs [128,12,12,32,8]
    const float* W  = (const float*)d_in[1];   // W      [4608,8,32]
    const float* dc = (const float*)d_in[2];   // digit_caps [10,8]
    float* out = (float*)d_out;                // 1280 + 80 floats
    float* ws  = (float*)d_ws;

    hipMemsetAsync(d_ws, 0, 1114 * sizeof(float), stream);
    digitcaps_votes<<<J_TOT / JPB, 256, 0, stream>>>(x, W, dc, ws);
    digitcaps_final<<<1, 128, 0, stream>>>(dc, ws, out);
}